// MyModel_61933428410907
// MI455X (gfx1250) — compile-verified
//
#include <hip/hip_runtime.h>

// Depthwise causal conv1d, K=3, f32.
//   y[n,c,l] = w[c,0]*x[n,c,l-2] + w[c,1]*x[n,c,l-1] + w[c,2]*x[n,c,l]
// (left zero-pad by 2). Bandwidth-bound: ~201 MB moved for 0.15 GFLOP
// (0.75 FLOP/byte) -> pure 128-bit streaming; WMMA deliberately not used.

#define D_MODEL 768
#define SEQ_L   4096
#define KTAPS   3
#define BLOCK   256
#define CHUNKS_PER_ROW (SEQ_L / 4)                  // 1024 float4 chunks
#define CHUNKS_PER_THREAD (CHUNKS_PER_ROW / BLOCK)  // 4

typedef float v4f __attribute__((ext_vector_type(4)));
typedef float v2f __attribute__((ext_vector_type(2)));

__global__ __launch_bounds__(BLOCK) void dwconv1d_causal_k3(
    const float* __restrict__ x,
    const float* __restrict__ w,
    float* __restrict__ y)
{
    const int row = blockIdx.x;            // 0 .. N*D_MODEL-1, one (n,c) row per block
    const int c   = row % D_MODEL;

    // Block-uniform taps -> one s_load_b96, kept in SGPRs.
    const float w0 = w[c * KTAPS + 0];
    const float w1 = w[c * KTAPS + 1];
    const float w2 = w[c * KTAPS + 2];

    const float* __restrict__ xr = x + (size_t)row * SEQ_L;
    float*       __restrict__ yr = y + (size_t)row * SEQ_L;

#pragma unroll
    for (int j = 0; j < CHUNKS_PER_THREAD; ++j) {
        const int chunk = threadIdx.x + j * BLOCK;   // 0..1023
        const int pos   = chunk * 4;                 // float index, 16B aligned

        // Prefetch next iteration's chunk into WGP$ (locality 3 -> WGP scope,
        // pulls into all cache levels). OOB prefetch past row end is dropped.
        __builtin_prefetch(xr + pos + BLOCK * 4, 0, 3);

        // Main 128-bit coalesced load.
        const v4f v = *reinterpret_cast<const v4f*>(xr + pos);

        // Causal neighbors x[pos-2], x[pos-1] as ONE unconditional b64 load
        // from a branchless, always-valid address (8B aligned). Only the
        // values are masked at pos==0 -> 2x v_cndmask, no exec-mask branches.
        const bool interior = (pos > 0);
        const int  npos     = interior ? (pos - 2) : 0;   // in-bounds always
        const v2f  nb       = *reinterpret_cast<const v2f*>(xr + npos);
        const float xm2 = interior ? nb.x : 0.0f;
        const float xm1 = interior ? nb.y : 0.0f;

        v4f o;
        o.x = __builtin_fmaf(w0, xm2, __builtin_fmaf(w1, xm1, w2 * v.x));
        o.y = __builtin_fmaf(w0, xm1, __builtin_fmaf(w1, v.x, w2 * v.y));
        o.z = __builtin_fmaf(w0, v.x, __builtin_fmaf(w1, v.y, w2 * v.z));
        o.w = __builtin_fmaf(w0, v.y, __builtin_fmaf(w1, v.z, w2 * v.w));

        // y is write-once, never re-read: non-temporal store (TH=NT) keeps it
        // from displacing the x stream in L2.
        __builtin_nontemporal_store(o, reinterpret_cast<v4f*>(yr + pos));
    }
}

extern "C" void kernel_launch(void* const* d_in, const int* in_sizes, int n_in,
                              void* d_out, int out_size, void* d_ws, size_t ws_size,
                              hipStream_t stream) {
    const float* x = (const float*)d_in[0];   // (8, 768, 4096) f32
    const float* w = (const float*)d_in[1];   // (768, 1, 3) f32
    float* y       = (float*)d_out;           // (8, 768, 4096) f32

    const int rows = in_sizes[0] / SEQ_L;     // 8 * 768 = 6144
    dwconv1d_causal_k3<<<rows, BLOCK, 0, stream>>>(x, w, y);
}